// RGB_47029891891283
// MI455X (gfx1250) — compile-verified
//
#include <hip/hip_runtime.h>
#include <cstdint>

#ifndef __has_builtin
#define __has_builtin(x) 0
#endif

#if __has_builtin(__builtin_amdgcn_global_load_async_to_lds_b128)
#define HAVE_ASYNC_BUILTIN 1
#else
#define HAVE_ASYNC_BUILTIN 0
#endif

// Hardware wait on ASYNCcnt plus a compiler memory barrier: the "memory"
// clobber stops LLVM from hoisting the LDS reads above the wait (the builtin
// alone does not write memory in IR terms, so loads could legally cross it).
#define WAIT_ASYNC(n) asm volatile("s_wait_asynccnt %0" ::"i"(n) : "memory")

#define WPB 8                   // waves per block (wave32)
#define BLOCK (WPB * 32)
#define RPW 16                  // rays per wave
#define NS 128                  // samples per ray
#define RGBF (NS * 3)           // 384 floats of rgb per ray
#define SLICE (RGBF + NS + NS)  // 640 floats: rgb | density | deltas

typedef __attribute__((ext_vector_type(4))) int v4i;
typedef __attribute__((address_space(1))) v4i glb_v4i;
typedef __attribute__((address_space(3))) v4i lds_v4i;

// Async DMA of 16 bytes/lane from global to LDS (ASYNCcnt-tracked).
__device__ __forceinline__ void async_copy16(const float* g, float* l) {
#if HAVE_ASYNC_BUILTIN
  __builtin_amdgcn_global_load_async_to_lds_b128((glb_v4i*)(float*)g,
                                                 (lds_v4i*)l, 0, 0);
#else
  unsigned loff =
      (unsigned)(unsigned long long)(__attribute__((address_space(3))) float*)l;
  asm volatile("global_load_async_to_lds_b128 %0, %1, off"
               ::"v"(loff), "v"(g)
               : "memory");
#endif
}

__global__ __launch_bounds__(BLOCK) void nerf_composite_kernel(
    const float* __restrict__ rgb, const float* __restrict__ density,
    const float* __restrict__ deltas, float* __restrict__ out, int nrays) {
  __shared__ float lds[WPB][2][SLICE];  // double-buffered per-wave slices
  const int lane = (int)(threadIdx.x & 31u);
  const int wv = (int)(threadIdx.x >> 5);
  const long long waveId = (long long)blockIdx.x * WPB + wv;
  const long long ray0 = waveId * (long long)RPW;
  if (ray0 >= nrays) return;  // wave-uniform

  // Issue the 5 async b128 loads (rgb x3, density, deltas) for ray r.
  auto issue = [&](long long r, float* slice) {
    const float* grow = rgb + r * (long long)RGBF;
    const float* gden = density + r * (long long)NS;
    const float* gdel = deltas + r * (long long)NS;
    float* lrgb = slice;
    float* lden = slice + RGBF;
    float* ldel = slice + RGBF + NS;
    async_copy16(grow + 4 * lane, lrgb + 4 * lane);
    async_copy16(grow + 4 * (lane + 32), lrgb + 4 * (lane + 32));
    async_copy16(grow + 4 * (lane + 64), lrgb + 4 * (lane + 64));
    async_copy16(gden + 4 * lane, lden + 4 * lane);
    async_copy16(gdel + 4 * lane, ldel + 4 * lane);
  };

  issue(ray0, lds[wv][0]);  // prime the pipeline

  for (int i = 0; i < RPW; ++i) {
    const long long r = ray0 + i;
    if (r >= nrays) break;  // wave-uniform
    const bool pf = (i + 1 < RPW) && (r + 1 < nrays);
    if (pf) {
      issue(r + 1, lds[wv][(i + 1) & 1]);
      WAIT_ASYNC(5);  // current ray's 5 loads done; next ray's 5 in flight
    } else {
      WAIT_ASYNC(0);
    }

    float* slice = lds[wv][i & 1];
    const float* lrgb = slice;
    const float* lden = slice + RGBF;
    const float* ldel = slice + RGBF + NS;

    // Lane L owns samples 4L..4L+3.
    float4 dv = *(const float4*)(lden + 4 * lane);
    float4 tv = *(const float4*)(ldel + 4 * lane);
    float dd0 = dv.x * tv.x, dd1 = dv.y * tv.y;
    float dd2 = dv.z * tv.z, dd3 = dv.w * tv.w;
    float p0 = dd0, p1 = p0 + dd1, p2 = p1 + dd2, p3 = p2 + dd3;

    // Wave32 inclusive scan of per-lane sums -> exclusive base per lane.
    float ssum = p3;
#pragma unroll
    for (int o = 1; o < 32; o <<= 1) {
      float t = __shfl_up(ssum, o);
      if (lane >= o) ssum += t;
    }
    float base = ssum - p3;

    // weights_j = exp(-excl_j) * (1 - exp(-dd_j)); advance T incrementally.
    float T = __expf(-base);
    float e0 = __expf(-dd0), e1 = __expf(-dd1);
    float e2 = __expf(-dd2), e3 = __expf(-dd3);
    float w0 = T * (1.f - e0); T *= e0;
    float w1 = T * (1.f - e1); T *= e1;
    float w2 = T * (1.f - e2); T *= e2;
    float w3 = T * (1.f - e3);

    // 12 rgb floats for this lane's 4 samples (3x ds_load_b128, 16B aligned).
    const float* lr = lrgb + 12 * lane;
    float4 c0 = *(const float4*)(lr + 0);
    float4 c1 = *(const float4*)(lr + 4);
    float4 c2 = *(const float4*)(lr + 8);

    float aR = w0 * c0.x + w1 * c0.w + w2 * c1.z + w3 * c2.y;
    float aG = w0 * c0.y + w1 * c1.x + w2 * c1.w + w3 * c2.z;
    float aB = w0 * c0.z + w1 * c1.y + w2 * c2.x + w3 * c2.w;

#pragma unroll
    for (int o = 16; o >= 1; o >>= 1) {
      aR += __shfl_xor(aR, o);
      aG += __shfl_xor(aG, o);
      aB += __shfl_xor(aB, o);
    }
    // All lanes now hold the full sums; lanes 0..2 store one component each.
    if (lane < 3) {
      float v = (lane == 0) ? aR : ((lane == 1) ? aG : aB);
      out[r * 3 + lane] = v;
    }
  }
}

extern "C" void kernel_launch(void* const* d_in, const int* in_sizes, int n_in,
                              void* d_out, int out_size, void* d_ws,
                              size_t ws_size, hipStream_t stream) {
  (void)n_in;
  (void)out_size;
  (void)d_ws;
  (void)ws_size;
  const float* rgb = (const float*)d_in[0];
  const float* den = (const float*)d_in[1];
  const float* del = (const float*)d_in[2];
  float* out = (float*)d_out;
  const int nrays = in_sizes[1] / NS;  // density is [rays, 128]
  const long long waves = ((long long)nrays + RPW - 1) / RPW;
  const int blocks = (int)((waves + WPB - 1) / WPB);
  nerf_composite_kernel<<<blocks, BLOCK, 0, stream>>>(rgb, den, del, out,
                                                      nrays);
}